// GAT_83794811945389
// MI455X (gfx1250) — compile-verified
//
#include <hip/hip_runtime.h>
#include <math.h>

// ---------------------------------------------------------------------------
// 2-layer GAT for MI455X (gfx1250, wave32).
//   L1: H1 = X @ W1  (WMMA f32 16x16x4, B panel staged in LDS via async loads)
//   L2: G2 = H2 @ W2 (same GEMM, W2 zero-padded 10 -> 16 cols)
// Edge softmax is computed unnormalized (distributive) with native f32 L2
// atomics; self-loops folded analytically into max-init + finalize.
// ---------------------------------------------------------------------------

#define NEG_SLOPE 0.2f
#define TILE_ROWS 128            // 8 wave32s * 16 rows per block

typedef __attribute__((ext_vector_type(2))) float v2f;
typedef __attribute__((ext_vector_type(8))) float v8f;

__device__ __forceinline__ float lrelu(float x) { return x > 0.f ? x : NEG_SLOPE * x; }

// float atomic max via native integer atomics (global_atomic_max_i32 / min_u32)
__device__ __forceinline__ void atomicMaxF(float* addr, float v) {
  if (v >= 0.f) atomicMax((int*)addr, __float_as_int(v));
  else          atomicMin((unsigned int*)addr, __float_as_uint(v));
}

// ---------------------------------------------------------------------------
// GEMM: H[M x Nc] = X[M x K] * W[K x ldw-span]; block = 128 rows x 16 cols.
// B panel (K x 16 floats) is async-copied to LDS once per block, then each
// wave runs K/4 fp32 WMMAs reading B fragments from LDS.
// A layout (f32 16x4): lanes0-15 rows, vgpr pair = K {0,1} lo-half / {2,3} hi.
// C/D: v8f, row = r + 8*laneHi, col = laneLo.
// Requires K % 64 == 0, K <= 256, ldw >= n0+16 (W pre-padded for Nc<16).
// ---------------------------------------------------------------------------
__global__ __launch_bounds__(256)
void gat_gemm_wmma_f32(const float* __restrict__ X, const float* __restrict__ W,
                       float* __restrict__ H, int M, int K, int ldw, int Nc,
                       int ldh, int nTiles) {
  __shared__ float Bs[256 * 16];                 // max K(=256) x 16 panel

  const int rb = blockIdx.x / nTiles;
  const int ct = blockIdx.x % nTiles;
  const int n0 = ct * 16;
  const int tid = threadIdx.x;

  // ---- async fill of B panel: 16 bytes per thread-iteration ----
  for (int f = tid * 4; f < K * 16; f += 256 * 4) {
    int k = f >> 4, c = f & 15;                  // 4 consecutive cols (c%4==0)
    const float* gsrc = W + (long)k * ldw + n0 + c;
    unsigned lds = (unsigned)(size_t)(&Bs[f]);   // low 32 bits = LDS offset
    asm volatile("global_load_async_to_lds_b128 %0, %1, off"
                 :: "v"(lds), "v"((unsigned long long)(size_t)gsrc)
                 : "memory");
  }
  asm volatile("s_wait_asynccnt 0" ::: "memory");
  __syncthreads();

  // ---- WMMA over K ----
  const int wv = tid >> 5, lane = tid & 31;
  const int laneLo = lane & 15, laneHi = lane >> 4;
  const int m0 = rb * TILE_ROWS + wv * 16;
  int mrow = m0 + laneLo; if (mrow > M - 1) mrow = M - 1;   // clamp tail reads
  const float* arow = X + (long)mrow * K;

  v8f c = {};
  for (int k0 = 0; k0 < K; k0 += 4) {
    int ka = k0 + 2 * laneHi;
    v2f a, b;
    a.x = arow[ka];
    a.y = arow[ka + 1];
    b.x = Bs[ka * 16 + laneLo];
    b.y = Bs[(ka + 1) * 16 + laneLo];
    // (neg_a, A, neg_b, B, c_mod, C, reuse_a, reuse_b)
    c = __builtin_amdgcn_wmma_f32_16x16x4_f32(false, a, false, b, (short)0, c,
                                              false, false);
  }

  int col = n0 + laneLo;
  if (col < Nc) {
    for (int r = 0; r < 8; ++r) {
      int row = m0 + r + 8 * laneHi;
      if (row < M) H[(long)row * ldh + col] = c[r];
    }
  }
}

// Zero-pad W [K x Nc] -> Wp [K x 16]
__global__ __launch_bounds__(256)
void gat_pad_w(const float* __restrict__ W, float* __restrict__ Wp, int K, int Nc) {
  int i = blockIdx.x * blockDim.x + threadIdx.x;
  if (i >= K * 16) return;
  int k = i >> 4, c = i & 15;
  Wp[i] = (c < Nc) ? W[k * Nc + c] : 0.f;
}

// ---------------------------------------------------------------------------
// Per-(node,head) attention coefficients; m initialized with self-loop value.
// ---------------------------------------------------------------------------
__global__ __launch_bounds__(256)
void gat_attn_coef(const float* __restrict__ H, const float* __restrict__ aS,
                   const float* __restrict__ aD, float* __restrict__ as,
                   float* __restrict__ ad, float* __restrict__ m,
                   int N, int Hn, int C) {
  int i = blockIdx.x * blockDim.x + threadIdx.x;
  if (i >= N * Hn) return;
  int n = i / Hn, h = i % Hn;
  const float* row = H + (long)n * Hn * C + (long)h * C;
  float ss = 0.f, sd = 0.f;
  for (int cidx = 0; cidx < C; ++cidx) {
    float v = row[cidx];
    ss += v * aS[h * C + cidx];
    sd += v * aD[h * C + cidx];
  }
  as[i] = ss;
  ad[i] = sd;
  m[i] = lrelu(ss + sd);          // self-loop seeds the running max
}

// ---------------------------------------------------------------------------
// Edge pass A: running max of leaky_relu(as[src] + ad[dst]) per (dst, head).
// ---------------------------------------------------------------------------
__global__ __launch_bounds__(256)
void gat_edge_max(const int* __restrict__ src, const int* __restrict__ dst,
                  const float* __restrict__ as, const float* __restrict__ ad,
                  float* __restrict__ m, int E, int Hn) {
  int e = blockIdx.x * blockDim.x + threadIdx.x;
  if (e >= E) return;
  int s = src[e], d = dst[e];
  for (int h = 0; h < Hn; ++h)
    atomicMaxF(&m[d * Hn + h], lrelu(as[s * Hn + h] + ad[d * Hn + h]));
}

// ---------------------------------------------------------------------------
// Edge pass B: one thread per (edge, head): w = exp(e - m[dst]);
// s[dst,h] += w; acc[dst,h,:] += w * H[src,h,:]  (native f32 add atomics).
// Gather rows with float4 when possible (L2-resident).
// ---------------------------------------------------------------------------
__global__ __launch_bounds__(256)
void gat_edge_agg(const int* __restrict__ src, const int* __restrict__ dst,
                  const float* __restrict__ as, const float* __restrict__ ad,
                  const float* __restrict__ m, const float* __restrict__ H,
                  float* __restrict__ ssum, float* __restrict__ acc,
                  int E, int Hn, int C) {
  long i = (long)blockIdx.x * blockDim.x + threadIdx.x;
  if (i >= (long)E * Hn) return;
  int e = (int)(i / Hn), h = (int)(i % Hn);
  int s = src[e], d = dst[e];
  float w = expf(lrelu(as[s * Hn + h] + ad[d * Hn + h]) - m[d * Hn + h]);
  unsafeAtomicAdd(&ssum[d * Hn + h], w);
  const float* hr = H + (long)s * Hn * C + (long)h * C;
  float* ar = acc + (long)d * Hn * C + (long)h * C;
  if ((C & 3) == 0) {
    const float4* h4 = (const float4*)hr;
    for (int q = 0; q < (C >> 2); ++q) {
      float4 v = h4[q];
      unsafeAtomicAdd(&ar[4 * q + 0], w * v.x);
      unsafeAtomicAdd(&ar[4 * q + 1], w * v.y);
      unsafeAtomicAdd(&ar[4 * q + 2], w * v.z);
      unsafeAtomicAdd(&ar[4 * q + 3], w * v.w);
    }
  } else {
    for (int cidx = 0; cidx < C; ++cidx)
      unsafeAtomicAdd(&ar[cidx], w * hr[cidx]);
  }
}

// ---------------------------------------------------------------------------
// Finalize: add self-loop term, normalize, add bias, optional ELU; in-place
// on acc (which becomes the next layer's input / the final output).
// ---------------------------------------------------------------------------
__global__ __launch_bounds__(256)
void gat_finalize(const float* __restrict__ H, const float* __restrict__ as,
                  const float* __restrict__ ad, const float* __restrict__ m,
                  const float* __restrict__ ssum, const float* __restrict__ b,
                  float* __restrict__ acc, int N, int Hn, int C, int applyElu) {
  int i = blockIdx.x * blockDim.x + threadIdx.x;
  if (i >= N * Hn) return;
  int n = i / Hn, h = i % Hn;
  float wse = expf(lrelu(as[i] + ad[i]) - m[i]);   // self-loop weight
  float inv = 1.f / (ssum[i] + wse + 1e-16f);
  const float* hr = H + (long)n * Hn * C + (long)h * C;
  float* ar = acc + (long)n * Hn * C + (long)h * C;
  for (int cidx = 0; cidx < C; ++cidx) {
    float v = (ar[cidx] + wse * hr[cidx]) * inv + b[h * C + cidx];
    if (applyElu) v = v > 0.f ? v : (expf(v) - 1.f);
    ar[cidx] = v;
  }
}

// ---------------------------------------------------------------------------
extern "C" void kernel_launch(void* const* d_in, const int* in_sizes, int n_in,
                              void* d_out, int out_size, void* d_ws, size_t ws_size,
                              hipStream_t stream) {
  const float* x   = (const float*)d_in[0];
  const int*   ei  = (const int*)d_in[1];     // [2,E]: src row then dst row
  const float* W1  = (const float*)d_in[2];
  const float* aS1 = (const float*)d_in[3];
  const float* aD1 = (const float*)d_in[4];
  const float* b1  = (const float*)d_in[5];
  const float* W2  = (const float*)d_in[6];
  const float* aS2 = (const float*)d_in[7];
  const float* aD2 = (const float*)d_in[8];
  const float* b2  = (const float*)d_in[9];

  const int HID   = in_sizes[5];            // 256
  const int IN    = in_sizes[2] / HID;      // 128
  const int N     = in_sizes[0] / IN;       // 50000
  const int E     = in_sizes[1] / 2;        // 800000
  const int HEADS = 8;
  const int C1    = HID / HEADS;            // 32
  const int OUTC  = in_sizes[9];            // 10

  const int* srcv = ei;
  const int* dstv = ei + E;

  // ---- workspace layout (floats) ----
  float* ws   = (float*)d_ws;
  float* H1   = ws;                 ws += (long)N * HID;   // X@W1
  float* acc1 = ws;                 ws += (long)N * HID;   // L1 accum -> H2
  float* as1  = ws;                 ws += (long)N * HEADS;
  float* ad1  = ws;                 ws += (long)N * HEADS;
  float* m1   = ws;                 ws += (long)N * HEADS;
  float* s1   = ws;                 ws += (long)N * HEADS;
  float* G2   = ws;                 ws += (long)N * OUTC;  // H2@W2
  float* as2  = ws;                 ws += (long)N;
  float* ad2  = ws;                 ws += (long)N;
  float* m2   = ws;                 ws += (long)N;
  float* s2   = ws;                 ws += (long)N;
  float* W2p  = ws;                 ws += (long)HID * 16;  // padded W2

  float* outp = (float*)d_out;      // also the layer-2 accumulator

  hipMemsetAsync(acc1, 0, (size_t)N * HID * sizeof(float), stream);
  hipMemsetAsync(s1,   0, (size_t)N * HEADS * sizeof(float), stream);
  hipMemsetAsync(s2,   0, (size_t)N * sizeof(float), stream);
  hipMemsetAsync(outp, 0, (size_t)out_size * sizeof(float), stream);

  const int BS = 256;               // 8 wave32s per block
  const int mBlocks = (N + TILE_ROWS - 1) / TILE_ROWS;

  // ---- Layer 1 ----
  {
    int nT = HID / 16;
    gat_gemm_wmma_f32<<<mBlocks * nT, BS, 0, stream>>>(
        x, W1, H1, N, IN, /*ldw=*/HID, /*Nc=*/HID, /*ldh=*/HID, nT);
  }
  gat_attn_coef<<<(N * HEADS + BS - 1) / BS, BS, 0, stream>>>(
      H1, aS1, aD1, as1, ad1, m1, N, HEADS, C1);
  gat_edge_max<<<(E + BS - 1) / BS, BS, 0, stream>>>(
      srcv, dstv, as1, ad1, m1, E, HEADS);
  {
    long work = (long)E * HEADS;
    gat_edge_agg<<<(int)((work + BS - 1) / BS), BS, 0, stream>>>(
        srcv, dstv, as1, ad1, m1, H1, s1, acc1, E, HEADS, C1);
  }
  gat_finalize<<<(N * HEADS + BS - 1) / BS, BS, 0, stream>>>(
      H1, as1, ad1, m1, s1, b1, acc1, N, HEADS, C1, /*elu=*/1);   // acc1 -> H2

  // ---- Layer 2 ----
  gat_pad_w<<<(HID * 16 + BS - 1) / BS, BS, 0, stream>>>(W2, W2p, HID, OUTC);
  gat_gemm_wmma_f32<<<mBlocks, BS, 0, stream>>>(
      acc1, W2p, G2, N, HID, /*ldw=*/16, /*Nc=*/OUTC, /*ldh=*/OUTC, 1);
  gat_attn_coef<<<(N + BS - 1) / BS, BS, 0, stream>>>(
      G2, aS2, aD2, as2, ad2, m2, N, 1, OUTC);
  gat_edge_max<<<(E + BS - 1) / BS, BS, 0, stream>>>(
      srcv, dstv, as2, ad2, m2, E, 1);
  gat_edge_agg<<<(E + BS - 1) / BS, BS, 0, stream>>>(
      srcv, dstv, as2, ad2, m2, G2, s2, outp, E, 1, OUTC);
  gat_finalize<<<(N + BS - 1) / BS, BS, 0, stream>>>(
      G2, as2, ad2, m2, s2, b2, outp, N, 1, OUTC, /*elu=*/0);
}